// PseudoLevinsonDurbinRecursion_5403068858942
// MI455X (gfx1250) — compile-verified
//
#include <hip/hip_runtime.h>
#include <hip/hip_bf16.h>

// PseudoLevinsonDurbinRecursion for MI455X (gfx1250, wave32).
// One wave per batch: Levinson-Durbin O(M^2) recursion (memory-bound optimal),
// followed by one WMMA-based Jacobi iterative-refinement step:
//   e = R*a + r1   (Toeplitz matvec via V_WMMA_F32_16X16X4_F32 tiles)
//   a <- a - e / r0
//   K = sqrt(r0 + r1 . a)

typedef __attribute__((ext_vector_type(2))) float v2f;
typedef __attribute__((ext_vector_type(8))) float v8f;

#define WPB 8     // waves per block
#define MM  64    // system size

__device__ __forceinline__ float wave_sum(float v) {
#pragma unroll
  for (int off = 16; off > 0; off >>= 1)
    v += __shfl_xor(v, off, 32);
  return v;
}

__global__ __launch_bounds__(256) void levinson_wmma_kernel(
    const float* __restrict__ rIn, float* __restrict__ out, int B) {
  __shared__ float rS[WPB][MM + 2];   // r[0..64]
  __shared__ float aS[WPB][MM];       // mirror of a for cross-lane access
  __shared__ float eS[WPB][MM];       // residual

  const int lane = threadIdx.x & 31;
  const int w    = threadIdx.x >> 5;
  const int b    = blockIdx.x * WPB + w;
  if (b >= B) return;  // wave-uniform exit

  // Coalesced load of r[0..64] (two 128B transactions + 1 scalar).
  const float* rg = rIn + (size_t)b * (MM + 1);
  rS[w][lane]      = rg[lane];
  rS[w][lane + 32] = rg[lane + 32];
  if (lane == 0) rS[w][MM] = rg[MM];

  // ---------------- Levinson-Durbin recursion ----------------
  // a[i] (0-based) solves R a = -r[1:], E tracks prediction error power.
  float aLo = 0.f, aHi = 0.f;
  aS[w][lane] = 0.f;
  aS[w][lane + 32] = 0.f;

  float E = rS[w][0];
#pragma unroll 1
  for (int m = 1; m <= MM; ++m) {
    // acc = r[m] + sum_{j=0}^{m-2} a[j] * r[m-1-j]
    float t = 0.f;
    if (lane <= m - 2)      t += aLo * rS[w][m - 1 - lane];
    if (lane + 32 <= m - 2) t += aHi * rS[w][m - 33 - lane];
    float acc = wave_sum(t) + rS[w][m];
    float k = -acc / E;

    // a[j] += k * a_old[m-2-j]; reads of the old mirror precede the writes
    // (same-wave LDS instructions are in-order).
    float rev   = (lane <= m - 2)      ? aS[w][m - 2 - lane]  : 0.f;
    float revHi = (lane + 32 <= m - 2) ? aS[w][m - 34 - lane] : 0.f;
    if (lane <= m - 2)      aLo += k * rev;
    if (lane + 32 <= m - 2) aHi += k * revHi;
    if (m - 1 < 32) { if (lane == m - 1)  aLo = k; }
    else            { if (lane == m - 33) aHi = k; }
    aS[w][lane]      = aLo;
    aS[w][lane + 32] = aHi;

    E *= (1.f - k * k);
  }

  // ---------------- WMMA residual: e = R*a + r1 ----------------
  // R is rebuilt on the fly from rS (R[i][j] = r[|i-j|]).
  // A tile (16x4 f32): lanes 0-15 hold K=0,1 ; lanes 16-31 hold K=2,3.
  // B tile (4x16 f32): column N=0 carries a[k..k+3]; other columns zero.
  // C/D (16x16 f32): column N=0 lives in lane 0 (M=0..7) and lane 16 (M=8..15).
  //
  // Loop order: kc outer (B tile loaded once, shared by 4 row-block WMMAs),
  // rb inner with 4 live accumulators (32 VGPRs). B-tile masking is done with
  // register selects (v_cndmask), NOT predicated loads, so EXEC stays all-ones
  // across the entire WMMA chain.
  const int  half = lane >> 4;
  const int  mrow = lane & 15;
  const bool col0 = (mrow == 0);

  v8f c[4];
#pragma unroll
  for (int rb = 0; rb < 4; ++rb) c[rb] = (v8f){0.f,0.f,0.f,0.f,0.f,0.f,0.f,0.f};

#pragma unroll 1
  for (int kc = 0; kc < 16; ++kc) {
    const int kA = kc * 4 + half * 2;
    const float bx = aS[w][kA];       // unconditional LDS read (cheap, no fault)
    const float by = aS[w][kA + 1];
    v2f B2;
    B2.x = col0 ? bx : 0.f;           // register select, no exec divergence
    B2.y = col0 ? by : 0.f;
#pragma unroll
    for (int rb = 0; rb < 4; ++rb) {
      const int row = rb * 16 + mrow;
      int d0 = row - kA;     d0 = (d0 < 0) ? -d0 : d0;
      int d1 = row - kA - 1; d1 = (d1 < 0) ? -d1 : d1;
      v2f A2;
      A2.x = rS[w][d0];
      A2.y = rS[w][d1];
      c[rb] = __builtin_amdgcn_wmma_f32_16x16x4_f32(
          /*neg_a=*/false, A2, /*neg_b=*/false, B2,
          /*c_mod=*/(short)0, c[rb], /*reuse_a=*/false, /*reuse_b=*/false);
    }
  }

  if (col0) {
#pragma unroll
    for (int rb = 0; rb < 4; ++rb) {
#pragma unroll
      for (int j = 0; j < 8; ++j) {
        const int i = rb * 16 + half * 8 + j;   // row index of y = R*a
        eS[w][i] = c[rb][j] + rS[w][i + 1];     // e = R*a + r1
      }
    }
  }

  // ---------------- Jacobi refinement + K ----------------
  // R ~ r0*I + small off-diagonals (autocorr of near-white input),
  // so one Jacobi step is contractive: a <- a - e/r0.
  const float inv_r0 = 1.f / rS[w][0];
  aLo -= eS[w][lane]      * inv_r0;
  aHi -= eS[w][lane + 32] * inv_r0;

  float td  = rS[w][lane + 1] * aLo + rS[w][lane + 33] * aHi;
  float dot = wave_sum(td);
  float Ksq = rS[w][0] + dot;
  float K   = sqrtf(fmaxf(Ksq, 0.f));

  float* og = out + (size_t)b * (MM + 1);
  if (lane == 0) og[0] = K;
  og[1 + lane]  = aLo;
  og[33 + lane] = aHi;
}

extern "C" void kernel_launch(void* const* d_in, const int* in_sizes, int n_in,
                              void* d_out, int out_size, void* d_ws, size_t ws_size,
                              hipStream_t stream) {
  const float* r = (const float*)d_in[0];
  float* out = (float*)d_out;
  const int B = in_sizes[0] / (MM + 1);     // 65536 batches of r[65]
  const int blocks = (B + WPB - 1) / WPB;   // one wave per batch, 8 waves/block
  levinson_wmma_kernel<<<blocks, 256, 0, stream>>>(r, out, B);
}